// ExpKernelMVHawkesProcess_63239098466633
// MI455X (gfx1250) — compile-verified
//
#include <hip/hip_runtime.h>
#include <hip/hip_bf16.h>

// Problem constants from the reference
#define BB 8
#define LL 2048
#define DD 10
#define MIN_INTENSITY 1e-12f

typedef __attribute__((ext_vector_type(2))) float v2f;
typedef __attribute__((ext_vector_type(8))) float v8f;

__device__ __forceinline__ float softplusf(float x) {
    // numerically stable softplus
    return (x > 20.0f) ? x : log1pf(__expf(x));
}

// ---------------------------------------------------------------------------
// Kernel 1: pairwise interaction sums via WMMA row-reduction.
// grid.x = BB * (LL/128); block = 256 threads (8 wave32s).
// Wave w of block handles the 16-row tile rt = (blockIdx%16)*8 + w of batch b.
// Writes logl_ws[b*LL + i] = valid_i ? log(max(mu[e_i] + sum_j inter, eps)) : 0
// ---------------------------------------------------------------------------
__global__ void __launch_bounds__(256)
hawkes_pairs(const float* __restrict__ tp,
             const int*   __restrict__ et,
             const float* __restrict__ log_alpha,
             const float* __restrict__ log_beta,
             const float* __restrict__ mu_raw,
             float*       __restrict__ logl_ws)
{
    __shared__ float2 colbuf[LL];       // {t_j, bitcast(e_j)} for this batch
    __shared__ float2 tab[DD * DD];     // {alpha*beta, beta}
    __shared__ float  mu_lds[DD];

    const int tid            = threadIdx.x;
    const int blocksPerBatch = LL / 128;          // 16
    const int b              = blockIdx.x / blocksPerBatch;
    const int rblk           = blockIdx.x % blocksPerBatch;
    const int lane           = tid & 31;

    // Stage column data for the whole batch (coalesced), plus tables.
    for (int idx = tid; idx < LL; idx += 256) {
        float2 c;
        c.x = tp[b * LL + idx];
        c.y = __int_as_float(et[b * LL + idx]);
        colbuf[idx] = c;
    }
    if (tid < DD * DD) {
        float al = softplusf(log_alpha[tid]);
        float be = softplusf(log_beta[tid]);
        float2 t2; t2.x = al * be; t2.y = be;
        tab[tid] = t2;
    }
    if (tid < DD) mu_lds[tid] = softplusf(mu_raw[tid]);
    __syncthreads();

    // This wave's 16-row tile: force scalar (wave-uniform) so the tile loop
    // uses scalar control flow instead of an exec-mask waterfall.
    const int rt   = __builtin_amdgcn_readfirstlane(rblk * 8 + (tid >> 5));
    const int i0   = rt * 16;
    const int m    = lane & 15;             // row within tile (A-matrix M)
    const int half = lane >> 4;             // K-half select (A-matrix layout)
    const int i    = i0 + m;

    const float ti = colbuf[i].x;
    const int   ei = __float_as_int(colbuf[i].y);
    const bool  vi = (ei >= 0);
    const int  eiD = (vi ? ei : 0) * DD;

    v8f acc = {};                           // C/D accumulator: row sums
    v2f bones; bones.x = 1.0f; bones.y = 1.0f;   // B = ones(4x16)

    // --- Full tiles: jt in [0, rt). Here j < i0 <= i always, so no triangle
    //     mask is needed; only validity masking.
    for (int jt = 0; jt < rt; ++jt) {
        const int j0 = jt * 16;
#pragma unroll
        for (int c = 0; c < 4; ++c) {
            v2f a;
#pragma unroll
            for (int v = 0; v < 2; ++v) {
                const int    j  = j0 + 4 * c + 2 * half + v;   // A[M=m, K=4c+2*half+v]
                const float2 cd = colbuf[j];                    // ds_load_b64
                const float  tj = cd.x;
                const int    ej = __float_as_int(cd.y);
                const bool   vj = (ej >= 0);
                const float2 ab = tab[eiD + (vj ? ej : 0)];     // {alpha*beta, beta}
                const float  td = ti - tj;
                const float  ex = fmaxf(-ab.y * td, -20.0f);    // clip(-b*td, -20, 0]
                const float val = ab.x * __expf(ex);
                a[v] = (vi && vj) ? val : 0.0f;
            }
            // D = A(16x4) * ones(4x16) + C  -> row sums accumulate in C
            acc = __builtin_amdgcn_wmma_f32_16x16x4_f32(
                false, a, false, bones, (short)0, acc, false, false);
        }
    }

    // --- Diagonal tile: jt == rt, apply strict lower-triangle mask j < i.
    {
        const int j0 = i0;
#pragma unroll
        for (int c = 0; c < 4; ++c) {
            v2f a;
#pragma unroll
            for (int v = 0; v < 2; ++v) {
                const int    j  = j0 + 4 * c + 2 * half + v;
                const float2 cd = colbuf[j];
                const float  tj = cd.x;
                const int    ej = __float_as_int(cd.y);
                const bool   vj = (ej >= 0);
                const float2 ab = tab[eiD + (vj ? ej : 0)];
                const float  td = ti - tj;
                const float  ex = fmaxf(-ab.y * td, -20.0f);
                const float val = ab.x * __expf(ex);
                a[v] = (vi && vj && (j < i)) ? val : 0.0f;
            }
            acc = __builtin_amdgcn_wmma_f32_16x16x4_f32(
                false, a, false, bones, (short)0, acc, false, false);
        }
    }

    // C/D layout: VGPR r holds row (r + 8*half) replicated across its 16 lanes.
    // Lanes 0 and 16 finalize 8 rows each.
    if ((lane & 15) == 0) {
#pragma unroll
        for (int r = 0; r < 8; ++r) {
            const int   row = i0 + r + 8 * half;
            const float s   = acc[r];
            const int   er  = __float_as_int(colbuf[row].y);
            float logl = 0.0f;
            if (er >= 0) {
                const float lam = mu_lds[er] + s;
                logl = logf(fmaxf(lam, MIN_INTENSITY));
            }
            logl_ws[b * LL + row] = logl;
        }
    }
}

// ---------------------------------------------------------------------------
// Kernel 2: per-batch finalize.  grid = BB blocks, 256 threads.
// out[b] = (nvalid ? sum_i logl : 0) - (sum_d mu_d * T_b + sum_j sum_d contrib)
// ---------------------------------------------------------------------------
__global__ void __launch_bounds__(256)
hawkes_final(const float* __restrict__ tp,
             const float* __restrict__ Tarr,
             const float* __restrict__ mu_raw,
             const float* __restrict__ log_alpha,
             const float* __restrict__ log_beta,
             const int*   __restrict__ et,
             const float* __restrict__ logl_ws,
             float*       __restrict__ out)
{
    __shared__ float a_lds[DD * DD];
    __shared__ float b_lds[DD * DD];
    __shared__ float mu_lds[DD];
    __shared__ float redL[256];
    __shared__ float redI[256];
    __shared__ int   redN[256];

    const int tid = threadIdx.x;
    const int b   = blockIdx.x;

    if (tid < DD * DD) {
        a_lds[tid] = softplusf(log_alpha[tid]);
        b_lds[tid] = softplusf(log_beta[tid]);
    }
    if (tid < DD) mu_lds[tid] = softplusf(mu_raw[tid]);
    __syncthreads();

    const float Tb = Tarr[b];

    float accL = 0.0f, accI = 0.0f;
    int   nv   = 0;
    for (int j = tid; j < LL; j += 256) {
        const int ej = et[b * LL + j];
        if (ej >= 0) {
            nv += 1;
            accL += logl_ws[b * LL + j];
            const float dt = Tb - tp[b * LL + j];
#pragma unroll
            for (int d = 0; d < DD; ++d) {
                const float be = b_lds[d * DD + ej];
                const float al = a_lds[d * DD + ej];
                accI += al * (1.0f - __expf(-be * dt));
            }
        }
    }

    redL[tid] = accL; redI[tid] = accI; redN[tid] = nv;
    __syncthreads();
    for (int s = 128; s > 0; s >>= 1) {
        if (tid < s) {
            redL[tid] += redL[tid + s];
            redI[tid] += redI[tid + s];
            redN[tid] += redN[tid + s];
        }
        __syncthreads();
    }

    if (tid == 0) {
        float musum = 0.0f;
#pragma unroll
        for (int d = 0; d < DD; ++d) musum += mu_lds[d];
        const float pos      = (redN[0] > 0) ? redL[0] : 0.0f;
        const float integral = musum * Tb + redI[0];
        out[b] = pos - integral;
    }
}

// ---------------------------------------------------------------------------
extern "C" void kernel_launch(void* const* d_in, const int* in_sizes, int n_in,
                              void* d_out, int out_size, void* d_ws, size_t ws_size,
                              hipStream_t stream)
{
    const float* tp        = (const float*)d_in[0];  // time_points (B,L)
    const float* T         = (const float*)d_in[1];  // T (B,)
    const float* mu_raw    = (const float*)d_in[2];  // (D,)
    const float* log_alpha = (const float*)d_in[3];  // (D,D)
    const float* log_beta  = (const float*)d_in[4];  // (D,D)
    const int*   et        = (const int*)d_in[5];    // event_types (B,L)
    float*       out       = (float*)d_out;          // (B,)
    float*       logl_ws   = (float*)d_ws;           // B*L floats = 64 KB

    hawkes_pairs<<<dim3(BB * (LL / 128)), dim3(256), 0, stream>>>(
        tp, et, log_alpha, log_beta, mu_raw, logl_ws);

    hawkes_final<<<dim3(BB), dim3(256), 0, stream>>>(
        tp, T, mu_raw, log_alpha, log_beta, et, logl_ws, out);
}